// MaskLayer_27496380629798
// MI455X (gfx1250) — compile-verified
//
#include <hip/hip_runtime.h>
#include <stdint.h>

// MaskLayer: per-(batch,channel) spatial-argmax L1 cone mask, applied to x.
// Memory-bound: 822 MB min traffic -> ~35us floor @ 23.3 TB/s.
// CDNA5 async global<->LDS DMA (ASYNCcnt) for all bulk data movement.
// 98 KB LDS tile per WG -> 3 workgroups co-resident per WGP (320 KB LDS),
// so DMA of one block overlaps compute of another. NT cache policy: pure
// streaming, every global byte touched exactly once.

#define IMG     14
#define POSN    196          // IMG*IMG
#define DEPTH   512
#define CHW     128          // channels handled per workgroup
#define THREADS 128
#define ROUNDS  ((POSN * CHW * 4) / (THREADS * 16))   // 49 b128 chunks / thread
#define CHUNKS_PER_POS ((CHW * 4) / 16)               // 32

#define BETA_OVER_IMG (4.0f / 14.0f)
#define TAU_F         (0.5f / 196.0f)

#define CPOL_NT 1            // temporal hint: non-temporal (streaming)

typedef int v4i __attribute__((ext_vector_type(4)));
typedef __attribute__((address_space(1))) v4i gv4i;   // global 16B chunk
typedef __attribute__((address_space(3))) v4i lv4i;   // LDS 16B chunk
typedef __attribute__((address_space(3))) void as3_void;

#if defined(__has_builtin)
#  if __has_builtin(__builtin_amdgcn_global_load_async_to_lds_b128)
#    define USE_ASYNC_LOAD_BUILTIN 1
#  endif
#  if __has_builtin(__builtin_amdgcn_global_store_async_from_lds_b128)
#    define USE_ASYNC_STORE_BUILTIN 1
#  endif
#endif

__device__ __forceinline__ void async_load_b128(const float* gsrc, float* ldst) {
#ifdef USE_ASYNC_LOAD_BUILTIN
    __builtin_amdgcn_global_load_async_to_lds_b128(
        (gv4i*)const_cast<float*>(gsrc), (lv4i*)ldst, 0, CPOL_NT);
#else
    uint32_t loff = (uint32_t)(uint64_t)(as3_void*)ldst;   // 32-bit LDS address
    asm volatile("global_load_async_to_lds_b128 %0, %1, off"
                 :: "v"(loff), "v"(gsrc) : "memory");
#endif
}

__device__ __forceinline__ void async_store_b128(float* gdst, const float* lsrc) {
#ifdef USE_ASYNC_STORE_BUILTIN
    __builtin_amdgcn_global_store_async_from_lds_b128(
        (gv4i*)gdst, (lv4i*)const_cast<float*>(lsrc), 0, CPOL_NT);
#else
    uint32_t loff = (uint32_t)(uint64_t)(as3_void*)lsrc;
    asm volatile("global_store_async_from_lds_b128 %0, %1, off"
                 :: "v"(gdst), "v"(loff) : "memory");
#endif
}

extern "C" __global__ __launch_bounds__(THREADS)
void mask_argmax_kernel(const float* __restrict__ x, float* __restrict__ out)
{
    extern __shared__ float tile[];               // POSN*CHW floats = 98 KB
    const int tid   = threadIdx.x;
    const int b     = blockIdx.x >> 2;            // batch
    const int cbase = (blockIdx.x & 3) * CHW;     // channel quarter

    const float* gsrc = x   + ((size_t)b * POSN) * DEPTH + cbase;
    float*       gdst = out + ((size_t)b * POSN) * DEPTH + cbase;

    // ---- Phase 1: async DMA global -> LDS (strided NHWC tile -> dense) ----
    // 32 chunks of 16B per spatial position: each wave covers exactly one
    // position's 512 contiguous bytes per round-slice (fully coalesced).
#pragma unroll
    for (int r = 0; r < ROUNDS; ++r) {
        int chunk = r * THREADS + tid;
        int pos   = chunk / CHUNKS_PER_POS;
        int lane  = chunk % CHUNKS_PER_POS;
        async_load_b128(gsrc + (size_t)pos * DEPTH + lane * 4, tile + chunk * 4);
    }
    asm volatile("s_wait_asynccnt 0" ::: "memory");
    __syncthreads();

    // ---- Phase 2: per-channel spatial max reduction + argmax ----
    float rowmax[IMG], colmax[IMG];
#pragma unroll
    for (int i = 0; i < IMG; ++i) {
        rowmax[i] = -__builtin_inff();
        colmax[i] = -__builtin_inff();
    }
#pragma unroll
    for (int p = 0; p < POSN; ++p) {              // h,w compile-time via unroll
        float v = tile[p * CHW + tid];            // conflict-free: lane == bank
        int h = p / IMG, w = p % IMG;
        rowmax[h] = fmaxf(rowmax[h], v);
        colmax[w] = fmaxf(colmax[w], v);
    }
    // First-occurrence argmax (strict '>' keeps the earliest max, as jnp.argmax)
    int ri = 0; float rb = rowmax[0];
#pragma unroll
    for (int i = 1; i < IMG; ++i) if (rowmax[i] > rb) { rb = rowmax[i]; ri = i; }
    int ci = 0; float cb = colmax[0];
#pragma unroll
    for (int i = 1; i < IMG; ++i) if (colmax[i] > cb) { cb = colmax[i]; ci = i; }

    // Apply mask in place in LDS: mask = TAU * max(1 - (BETA/IMG)*L1, -1)
#pragma unroll
    for (int p = 0; p < POSN; ++p) {
        int h = p / IMG, w = p % IMG;
        int dh = h - ri; dh = dh < 0 ? -dh : dh;
        int dw = w - ci; dw = dw < 0 ? -dw : dw;
        float val = 1.0f - BETA_OVER_IMG * (float)(dh + dw);
        float m   = TAU_F * fmaxf(val, -1.0f);
        int idx = p * CHW + tid;
        tile[idx] = m * tile[idx];
    }
    __syncthreads();

    // ---- Phase 3: async DMA LDS -> global (b128 coalesced NT stores) ----
#pragma unroll
    for (int r = 0; r < ROUNDS; ++r) {
        int chunk = r * THREADS + tid;
        int pos   = chunk / CHUNKS_PER_POS;
        int lane  = chunk % CHUNKS_PER_POS;
        async_store_b128(gdst + (size_t)pos * DEPTH + lane * 4, tile + chunk * 4);
    }
    asm volatile("s_wait_asynccnt 0" ::: "memory");
}

extern "C" void kernel_launch(void* const* d_in, const int* in_sizes, int n_in,
                              void* d_out, int out_size, void* d_ws, size_t ws_size,
                              hipStream_t stream) {
    (void)n_in; (void)d_ws; (void)ws_size; (void)out_size;
    const float* x  = (const float*)d_in[0];
    float*      out = (float*)d_out;

    const int B = in_sizes[0] / (POSN * DEPTH);       // 1024
    const int quarters = DEPTH / CHW;                 // 4
    dim3 grid(B * quarters), block(THREADS);
    size_t ldsBytes = (size_t)POSN * CHW * sizeof(float);   // 98 KB dynamic LDS

    mask_argmax_kernel<<<grid, block, ldsBytes, stream>>>(x, out);
}